// ModulatedDeformConv_8014408975113
// MI455X (gfx1250) — compile-verified
//
#include <hip/hip_runtime.h>
#include <hip/hip_bf16.h>

typedef __attribute__((ext_vector_type(2))) float v2f;
typedef __attribute__((ext_vector_type(8))) float v8f;

#define B_    4
#define CIN   64
#define H_    128
#define W_    128
#define COUT  64
#define K2    9
#define HW    (H_ * W_)
#define KDIM  (CIN * K2)   // 576
#define TILE_P 64          // output positions per workgroup
#define LDSP   65          // padded LDS row (65*9=585 ≡ 9 mod 64 -> conflict-free writes)

static __device__ __forceinline__ int clampi(int v, int lo, int hi) {
    return v < lo ? lo : (v > hi ? hi : v);
}

// ---------------------------------------------------------------------------
// Pass 1: NCHW -> NHWC transpose of x so bilinear corner gathers are coalesced
// across the channel dimension (32 lanes = 32 consecutive floats = 128B).
// ---------------------------------------------------------------------------
__global__ __launch_bounds__(256) void nchw_to_nhwc(const float* __restrict__ x,
                                                    float* __restrict__ xT) {
    __shared__ float tile[32][33];
    const int bw = blockIdx.x;            // W/32 = 4
    const int bc = blockIdx.y;            // C/32 = 2
    const int bh = blockIdx.z;            // B*H = 512
    const int b  = bh >> 7;
    const int h  = bh & 127;
    const int tx = threadIdx.x & 31;
    const int ty = threadIdx.x >> 5;      // 0..7
#pragma unroll
    for (int j = 0; j < 4; ++j) {
        const int c = bc * 32 + ty + j * 8;
        tile[ty + j * 8][tx] = x[(((size_t)(b * CIN + c) * H_ + h) * W_) + bw * 32 + tx];
    }
    __syncthreads();
#pragma unroll
    for (int j = 0; j < 4; ++j) {
        const int w = bw * 32 + ty + j * 8;
        xT[((size_t)(b * H_ + h) * W_ + w) * CIN + bc * 32 + tx] = tile[tx][ty + j * 8];
    }
}

// ---------------------------------------------------------------------------
// Pass 2: fused deformable sampling (-> LDS column tile) + fp32 WMMA GEMM.
// One WG = (batch b, output row ho, 64-wide Wo tile). 256 threads = 8 waves.
// ---------------------------------------------------------------------------
__global__ __launch_bounds__(256) void dcnv2_fused(
    const float* __restrict__ xT,       // [B,H,W,C]
    const float* __restrict__ offset,   // [B,2*K2,H,W]  (interleaved y,x per tap)
    const float* __restrict__ mask,     // [B,K2,H,W]
    const float* __restrict__ weight,   // [COUT, KDIM]  (c*9 + tap, matches reference reshape)
    const float* __restrict__ bias,     // [COUT]
    float* __restrict__ out)            // [B,COUT,H,W]
{
    extern __shared__ float cols[];     // [KDIM][LDSP] fp32 column tile
    const int wt  = blockIdx.x;         // 0..1 (Wo tile)
    const int ho  = blockIdx.y;         // 0..127
    const int b   = blockIdx.z;         // 0..3
    const int tid = threadIdx.x;

    // ---------------- Stage 1: deformable bilinear sampling into LDS --------
    // Lane -> channel c (coalesced NHWC gathers); each wave shares one
    // (tap, position) pair per iteration, so offset/mask loads broadcast.
    {
        const int c      = tid & 63;
        const int stream = tid >> 6;    // 0..3 pair streams
        const float* xb  = xT + (size_t)b * HW * CIN + c;
        for (int pair = stream; pair < K2 * TILE_P; pair += 4) {
            const int tap = pair >> 6;          // 0..8
            const int p   = pair & 63;          // position within tile
            const int wo  = wt * TILE_P + p;

            const float dy = offset[((size_t)(b * 2 * K2 + 2 * tap    ) * H_ + ho) * W_ + wo];
            const float dx = offset[((size_t)(b * 2 * K2 + 2 * tap + 1) * H_ + ho) * W_ + wo];
            const float mv = mask  [((size_t)(b * K2 + tap            ) * H_ + ho) * W_ + wo];

            const float py = (float)(ho - 1 + tap / 3) + dy;
            const float px = (float)(wo - 1 + tap % 3) + dx;
            const float fy = floorf(py), fx = floorf(px);
            const int   y0 = (int)fy,    x0 = (int)fx;
            const float ly = py - fy,    lx = px - fx;
            const int   y1 = y0 + 1,     x1 = x0 + 1;

            const int y0c = clampi(y0, 0, H_ - 1), x0c = clampi(x0, 0, W_ - 1);
            const int y1c = clampi(y1, 0, H_ - 1), x1c = clampi(x1, 0, W_ - 1);
            const bool vy0 = (y0 >= 0) & (y0 < H_), vx0 = (x0 >= 0) & (x0 < W_);
            const bool vy1 = (y1 >= 0) & (y1 < H_), vx1 = (x1 >= 0) & (x1 < W_);

            const float w00 = (vy0 & vx0) ? (1.f - ly) * (1.f - lx) : 0.f;
            const float w01 = (vy0 & vx1) ? (1.f - ly) * lx         : 0.f;
            const float w10 = (vy1 & vx0) ? ly * (1.f - lx)         : 0.f;
            const float w11 = (vy1 & vx1) ? ly * lx                 : 0.f;

            float v = w00 * xb[((size_t)(y0c * W_) + x0c) * CIN]
                    + w01 * xb[((size_t)(y0c * W_) + x1c) * CIN]
                    + w10 * xb[((size_t)(y1c * W_) + x0c) * CIN]
                    + w11 * xb[((size_t)(y1c * W_) + x1c) * CIN];

            cols[(c * K2 + tap) * LDSP + p] = v * mv;
        }
    }
    __syncthreads();

    // ---------------- Stage 2: fp32 WMMA GEMM over the LDS tile -------------
    // D[64 x 64] = W[64 x 576] * cols[576 x 64]; 16 tiles of 16x16,
    // 8 waves x 2 M-tiles each (shared B fragment), K-loop of 144 x (K=4).
    const int lane  = tid & 31;
    const int wid   = tid >> 5;            // 0..7
    const int nt    = wid & 3;             // N tile
    const int mt0   = (wid >> 2) * 2;      // M tiles {mt0, mt0+1}
    const int mrow  = lane & 15;
    const int khalf = lane >> 4;           // A/B layout: lanes 16..31 hold K=2,3
    const int n     = nt * 16 + (lane & 15);

    const float* Wa = weight + (size_t)(mt0 * 16 + mrow) * KDIM + 2 * khalf;
    const float* Wb = Wa + (size_t)16 * KDIM;

    v8f acc0 = {};
    v8f acc1 = {};
    for (int k = 0; k < KDIM; k += 4) {
        const v2f a0 = *(const v2f*)(Wa + k);   // A: lane m, K = 2*khalf + {0,1}
        const v2f a1 = *(const v2f*)(Wb + k);
        v2f bf;
        bf.x = cols[(k + 2 * khalf    ) * LDSP + n];  // B: lane n, K = 2*khalf + {0,1}
        bf.y = cols[(k + 2 * khalf + 1) * LDSP + n];
        acc0 = __builtin_amdgcn_wmma_f32_16x16x4_f32(false, a0, false, bf,
                                                     (short)0, acc0, false, false);
        acc1 = __builtin_amdgcn_wmma_f32_16x16x4_f32(false, a1, false, bf,
                                                     (short)0, acc1, false, false);
    }

    // ---------------- Store: C/D layout -> NCHW output + bias ---------------
    const int mhalf = lane >> 4;           // lanes 16..31 hold rows M+8
    const int woAbs = wt * TILE_P + nt * 16 + (lane & 15);
#pragma unroll
    for (int r = 0; r < 8; ++r) {
        const int m0 = mt0 * 16 + r + 8 * mhalf;
        const int m1 = m0 + 16;
        out[(size_t)(b * COUT + m0) * HW + ho * W_ + woAbs] = acc0[r] + bias[m0];
        out[(size_t)(b * COUT + m1) * HW + ho * W_ + woAbs] = acc1[r] + bias[m1];
    }
}

// ---------------------------------------------------------------------------
extern "C" void kernel_launch(void* const* d_in, const int* in_sizes, int n_in,
                              void* d_out, int out_size, void* d_ws, size_t ws_size,
                              hipStream_t stream) {
    const float* x      = (const float*)d_in[0];
    const float* offset = (const float*)d_in[1];
    const float* mask   = (const float*)d_in[2];
    const float* weight = (const float*)d_in[3];
    const float* bias   = (const float*)d_in[4];
    float* out = (float*)d_out;
    float* xT  = (float*)d_ws;   // [B,H,W,C] fp32 = 16.8 MB

    dim3 g1(W_ / 32, CIN / 32, B_ * H_);
    nchw_to_nhwc<<<g1, 256, 0, stream>>>(x, xT);

    dim3 g2(W_ / TILE_P, H_, B_);
    const size_t ldsBytes = (size_t)KDIM * LDSP * sizeof(float);   // 149,760 B
    dcnv2_fused<<<g2, 256, ldsBytes, stream>>>(xT, offset, mask, weight, bias, out);
}